// SSIMLoss_2946347565673
// MI455X (gfx1250) — compile-verified
//
#include <hip/hip_runtime.h>

typedef float v2f __attribute__((ext_vector_type(2)));
typedef float v8f __attribute__((ext_vector_type(8)));
typedef unsigned int v4u __attribute__((ext_vector_type(4)));
typedef unsigned int v8u __attribute__((ext_vector_type(8)));

#define Cc 8
#define Hh 2048
#define Ww 2048
#define STRIP_ROWS 64
#define TILES_X  (Ww / 16)                   // 128
#define STRIPS_Y (Hh / STRIP_ROWS)           // 32
#define NBLOCKS  (Cc * STRIPS_Y * TILES_X)   // 32768

__global__ void ssim_zero_kernel(float* out) { out[0] = 0.0f; }

// guarded 1-D weight lookup: taps outside [0,10] are structural zeros of the band matrix
__device__ __forceinline__ float wsel(const float* w, int d) {
    return (d >= 0 && d < 11) ? w[d] : 0.0f;
}

__global__ __launch_bounds__(32) void ssim_wmma_kernel(
    const float* __restrict__ x, const float* __restrict__ y,
    const float* __restrict__ kern, float* __restrict__ out)
{
    // one wave per block; LDS = 42,368 bytes
    __shared__ float wsA[16];           // vertical (row-sum) weights a[i]
    __shared__ float wsB[16];           // horizontal profile b[j]
    __shared__ float XT[80 * 28];       // x halo strip: 80 rows x 28 cols
    __shared__ float YT[80 * 28];       // y halo strip
    __shared__ float HB[5 * 76 * 16];   // horizontal-pass results: 5 fields x 76 rows x 16 cols

    const int tid  = threadIdx.x;
    const int lane = tid & 31;
    const int lo   = lane & 15;         // M/N index within fragment
    const int hi   = lane >> 4;         // lane-half selector (per ISA fp32 WMMA layouts)

    const int blk  = blockIdx.x;
    const int ch   = blk / (STRIPS_Y * TILES_X);
    const int brem = blk % (STRIPS_Y * TILES_X);
    const int row0 = (brem / TILES_X) * STRIP_ROWS;   // strip top output row
    const int col0 = (brem % TILES_X) * 16;           // output col

    const size_t chbase = (size_t)ch * Hh * Ww;

    // ---- stage 80x28 halo strips (zero padded like the reference conv) into LDS ----
    const bool interior = (row0 > 0) && (col0 > 0);   // top-left origin in-bounds
    if (interior) {
        // Tensor Data Mover path (CDNA5 ISA D# groups 0/1).
        // tile = 28 cols x 80 rows of 4B elements starting at (row0-5, col0-5);
        // tensor_dim = remaining extent so bottom/right OOB rows/cols are written as zeros;
        // tile_dim0 == LDS row stride (28) -> rows pack contiguously, no pad needed.
        const unsigned tdim0 = (unsigned)(Ww - (col0 - 5));
        const unsigned tdim1 = (unsigned)(Hh - (row0 - 5));
        v8u g1;
        g1[0] = 0x00020000u;                              // workgroup_mask=0, data_size=2 (4B)
        g1[1] = (tdim0 & 0xFFFFu) << 16;                  // tensor_dim0[15:0]
        g1[2] = (tdim0 >> 16) | ((tdim1 & 0xFFFFu) << 16);// tensor_dim0[31:16] | tensor_dim1[15:0]
        g1[3] = (tdim1 >> 16) | (28u << 16);              // tensor_dim1[31:16] | tile_dim0=28
        g1[4] = 80u;                                      // tile_dim1=80, tile_dim2=0
        g1[5] = (unsigned)Ww;                             // tensor_dim0_stride[31:0]
        g1[6] = 0u;                                       // stride hi, tensor_dim1_stride lo
        g1[7] = 0u;

        const unsigned long long gx =
            (unsigned long long)(uintptr_t)(x + chbase + (size_t)(row0 - 5) * Ww + (col0 - 5));
        const unsigned long long gy =
            (unsigned long long)(uintptr_t)(y + chbase + (size_t)(row0 - 5) * Ww + (col0 - 5));
        const unsigned ldsx = (unsigned)(uintptr_t)(void*)XT;  // low 32b of flat LDS addr = LDS offset
        const unsigned ldsy = (unsigned)(uintptr_t)(void*)YT;

        v4u g0x, g0y;
        g0x[0] = 1u;                     // count=1, no gather
        g0x[1] = ldsx;                   // lds_addr
        g0x[2] = (unsigned)gx;           // global_addr[31:0]
        g0x[3] = (unsigned)((gx >> 32) & 0x01FFFFFFu) | 0x80000000u;  // addr[56:32] | type=2
        g0y[0] = 1u;
        g0y[1] = ldsy;
        g0y[2] = (unsigned)gy;
        g0y[3] = (unsigned)((gy >> 32) & 0x01FFFFFFu) | 0x80000000u;

        asm volatile("tensor_load_to_lds %0, %1" :: "s"(g0x), "s"(g1) : "memory");
        asm volatile("tensor_load_to_lds %0, %1" :: "s"(g0y), "s"(g1) : "memory");
        __builtin_amdgcn_s_wait_tensorcnt(0);
    } else {
        // edge strips: manual zero-padded fill (handles negative start offsets)
        for (int i = lane; i < 80 * 28; i += 32) {
            const int r = i / 28, c = i % 28;
            const int gr = row0 - 5 + r, gc = col0 - 5 + c;
            const bool ok = (gr >= 0) & (gr < Hh) & (gc >= 0) & (gc < Ww);
            const size_t gi = chbase + (size_t)(ok ? gr : 0) * Ww + (size_t)(ok ? gc : 0);
            XT[i] = ok ? x[gi] : 0.0f;
            YT[i] = ok ? y[gi] : 0.0f;
        }
    }

    // ---- separable factors from the rank-1 11x11 kernel (channel 0) ----
    if (tid < 11) {
        float s = 0.0f;
        #pragma unroll
        for (int j = 0; j < 11; ++j) s += kern[tid * 11 + j];
        wsA[tid] = s;                                  // a[i] = row sums
    }
    __syncthreads();
    if (tid < 11) wsB[tid] = kern[5 * 11 + tid] / wsA[5];  // b[j] = center row / its sum
    __syncthreads();

    // ---- per-lane weight fragments for the band matmuls ----
    // A fp32 16x4 layout: lane holds (M=lane%16, K = 2*hi + {0,1}) in {.x,.y}
    // B fp32 4x16 layout (row-striped): comp0 = row 2*hi, comp1 = row 2*hi+1, N = lane%16
    v2f wBh[7], wAv[7];
    #pragma unroll
    for (int t = 0; t < 7; ++t) {
        const int k0 = 4 * t + 2 * hi;
        wBh[t].x = wsel(wsB, k0 - lo);       // Wb[c][n] = b[c-n]
        wBh[t].y = wsel(wsB, k0 + 1 - lo);
        wAv[t].x = wsel(wsA, k0 - lo);       // Wa[m][k] = a[k-m]
        wAv[t].y = wsel(wsA, k0 + 1 - lo);
    }

    // ---- horizontal pass over the whole strip: H(74x16) = Field(80x28) x Wb(28x16) ----
    for (int g = 0; g < 5; ++g) {            // 5 row groups cover rows 0..79 exactly
        const int r = lo + 16 * g;
        float xv[14], yv[14];
        #pragma unroll
        for (int t = 0; t < 7; ++t) {        // load inputs once, reuse for all 5 fields
            const int c0 = 4 * t + 2 * hi;
            xv[2*t]   = XT[r * 28 + c0];
            xv[2*t+1] = XT[r * 28 + c0 + 1];
            yv[2*t]   = YT[r * 28 + c0];
            yv[2*t+1] = YT[r * 28 + c0 + 1];
        }
        #pragma unroll
        for (int f = 0; f < 5; ++f) {
            v8f acc = {};
            #pragma unroll
            for (int t = 0; t < 7; ++t) {
                const float xa = xv[2*t], xb = xv[2*t+1];
                const float ya = yv[2*t], yb = yv[2*t+1];
                v2f A;
                switch (f) {
                    case 0:  A.x = xa;      A.y = xb;      break;  // x
                    case 1:  A.x = ya;      A.y = yb;      break;  // y
                    case 2:  A.x = xa * xa; A.y = xb * xb; break;  // x*x
                    case 3:  A.x = ya * ya; A.y = yb * yb; break;  // y*y
                    default: A.x = xa * ya; A.y = xb * yb; break;  // x*y
                }
                acc = __builtin_amdgcn_wmma_f32_16x16x4_f32(
                          false, A, false, wBh[t], (short)0, acc, false, false);
            }
            // D layout: element (row = v + 8*hi, col = lo); keep rows 0..73, zero 74..75
            #pragma unroll
            for (int v = 0; v < 8; ++v) {
                const int rr = 16 * g + v + 8 * hi;
                if (rr < 74)      HB[f * 1216 + rr * 16 + lo] = acc[v];
                else if (rr < 76) HB[f * 1216 + rr * 16 + lo] = 0.0f;
            }
        }
    }
    __syncthreads();

    // ---- vertical pass per 16-row tile: Out(16x16) = Wa(16x28) x H(28x16) ----
    float ssum = 0.0f;
    const float C1 = 1e-4f, C2 = 9e-4f;
    for (int tt = 0; tt < 4; ++tt) {
        v8f cv[5];
        #pragma unroll
        for (int f = 0; f < 5; ++f) {
            v8f acc = {};
            #pragma unroll
            for (int t = 0; t < 7; ++t) {
                const int k0 = 16 * tt + 4 * t + 2 * hi;
                v2f B;
                B.x = HB[f * 1216 + k0 * 16 + lo];
                B.y = HB[f * 1216 + (k0 + 1) * 16 + lo];
                acc = __builtin_amdgcn_wmma_f32_16x16x4_f32(
                          false, wAv[t], false, B, (short)0, acc, false, false);
            }
            cv[f] = acc;
        }
        #pragma unroll
        for (int v = 0; v < 8; ++v) {
            const float mx  = cv[0][v], my  = cv[1][v];
            const float mxx = cv[2][v], myy = cv[3][v], mxy = cv[4][v];
            const float mx2 = mx * mx, my2 = my * my, mxyv = mx * my;
            const float sx  = mxx - mx2, sy = myy - my2, sxy = mxy - mxyv;
            const float num = (2.0f * mxyv + C1) * (2.0f * sxy + C2);
            const float den = (mx2 + my2 + C1) * (sx + sy + C2);
            ssum += num / den;
        }
    }

    // ---- wave reduction + global mean accumulation ----
    #pragma unroll
    for (int off = 16; off > 0; off >>= 1) ssum += __shfl_xor(ssum, off, 32);
    if (lane == 0) atomicAdd(out, ssum * (1.0f / 33554432.0f));
}

extern "C" void kernel_launch(void* const* d_in, const int* in_sizes, int n_in,
                              void* d_out, int out_size, void* d_ws, size_t ws_size,
                              hipStream_t stream) {
    (void)in_sizes; (void)n_in; (void)out_size; (void)d_ws; (void)ws_size;
    const float* x = (const float*)d_in[0];
    const float* y = (const float*)d_in[1];
    const float* k = (const float*)d_in[2];
    float* out = (float*)d_out;

    ssim_zero_kernel<<<dim3(1), dim3(1), 0, stream>>>(out);
    ssim_wmma_kernel<<<dim3(NBLOCKS), dim3(32), 0, stream>>>(x, y, k, out);
}